// RegionLayer_9500467658862
// MI455X (gfx1250) — compile-verified
//
#include <hip/hip_runtime.h>
#include <math.h>

typedef __attribute__((ext_vector_type(2))) float v2f;
typedef __attribute__((ext_vector_type(8))) float v8f;

#define GDIM   19
#define NA     5
#define NB     32
#define NC     20
#define NCH    25          // NC + 5
#define GG     361         // GDIM*GDIM
#define NCELL  57760       // NB*NA*GG
#define NT     256         // number of targets
#define NBLK   226         // ceil(NCELL/256)

__constant__ float c_AW[NA] = {1.3221f, 3.19275f, 5.05587f, 9.47112f, 11.2364f};
__constant__ float c_AH[NA] = {1.73145f, 4.00944f, 8.09892f, 4.84053f, 10.0071f};

__device__ __forceinline__ float sigf(float v) { return 1.0f / (1.0f + expf(-v)); }

// ---- wave32 sum via V_WMMA_F32_16X16X4_F32 -------------------------------
// A layout (16x4 f32): VGPR0 lanes0-15 -> K=0, lanes16-31 -> K=2 (VGPR1 = 0).
// With B = ones: D[m,n] = v_m + v_{m+16}.  D layout: lane L holds 8 rows of
// column (L&15), so sum(c[0..7]) on lanes 0-15 covers rows 0-7, lanes 16-31
// rows 8-15; one xor-16 shuffle completes the exact f32 wave sum.
__device__ __forceinline__ float wave_sum_wmma(float v) {
    v2f a; a[0] = v;    a[1] = 0.0f;
    v2f b; b[0] = 1.0f; b[1] = 1.0f;
    v8f c = {};
    c = __builtin_amdgcn_wmma_f32_16x16x4_f32(false, a, false, b,
                                              (short)0, c, false, false);
    float s = c[0] + c[1] + c[2] + c[3] + c[4] + c[5] + c[6] + c[7];
    s += __shfl_xor(s, 16, 32);
    return s;
}

// Block-wide sum (256 threads = 8 waves). Valid on thread 0. All threads
// must call (keeps EXEC all-ones for the WMMA).
__device__ __forceinline__ float block_sum(float v, float* sbuf) {
    float w = wave_sum_wmma(v);
    int lane = threadIdx.x & 31, wid = threadIdx.x >> 5;
    if (lane == 0) sbuf[wid] = w;
    __syncthreads();
    float r = 0.0f;
    if (threadIdx.x == 0)
        for (int i = 0; i < 8; ++i) r += sbuf[i];
    __syncthreads();
    return r;
}

// ---- kernel 0: zero the target mask --------------------------------------
__global__ void k_zero_mask(int* mask) {
    int i = blockIdx.x * blockDim.x + threadIdx.x;
    if (i < NCELL) mask[i] = 0;
}

// ---- kernel 1: per-target work (one block, 256 threads) ------------------
// Writes: mask bits, target box corners/area (grid units) into ws,
// scal[0..2] = loss_cls, loss_obj, loss_coords (already scaled).
__global__ void k_targets(const float* __restrict__ x,
                          const float* __restrict__ tg,
                          int* __restrict__ mask,
                          float* __restrict__ tbox,   // SoA: x1,y1,x2,y2,area
                          float* __restrict__ scal) {
    __shared__ float red[8];
    int t = threadIdx.x;

    float si_f = tg[t * 6 + 0], ci_f = tg[t * 6 + 1];
    float tgx = tg[t * 6 + 2] * (float)GDIM;
    float tgy = tg[t * 6 + 3] * (float)GDIM;
    float tgw = tg[t * 6 + 4] * (float)GDIM;
    float tgh = tg[t * 6 + 5] * (float)GDIM;
    int si = (int)si_f, ci = (int)ci_f;
    int gx = (int)tgx, gy = (int)tgy;

    // anchor argmax (strict > == first-max tie-break, matches jnp.argmax)
    int ai = 0; float best = -1.0f;
    for (int k = 0; k < NA; ++k) {
        float inter = fminf(c_AW[k], tgw) * fminf(c_AH[k], tgh);
        float uni   = c_AW[k] * c_AH[k] + tgw * tgh - inter;
        float iou   = inter / uni;
        if (iou > best) { best = iou; ai = k; }
    }

    mask[((si * NA + ai) * GDIM + gy) * GDIM + gx] = 1;

    // target box corners for the noobj pass
    tbox[0 * NT + t] = tgx - tgw * 0.5f;
    tbox[1 * NT + t] = tgy - tgh * 0.5f;
    tbox[2 * NT + t] = tgx + tgw * 0.5f;
    tbox[3 * NT + t] = tgy + tgh * 0.5f;
    tbox[4 * NT + t] = tgw * tgh;

    // gather prediction at the target cell
    int base = (si * NA * NCH + ai * NCH) * GG + gy * GDIM + gx;
    float txv = x[base + 0 * GG], tyv = x[base + 1 * GG];
    float twv = x[base + 2 * GG], thv = x[base + 3 * GG];
    float cfv = x[base + 4 * GG];

    float sx = sigf(txv), sy = sigf(tyv);
    float pw = expf(twv) * c_AW[ai], ph = expf(thv) * c_AH[ai];
    float px = sx + (float)gx, py = sy + (float)gy;

    // iou(pred, target), center format, grid units
    float px1 = px - pw * 0.5f, px2 = px + pw * 0.5f;
    float py1 = py - ph * 0.5f, py2 = py + ph * 0.5f;
    float qx1 = tgx - tgw * 0.5f, qx2 = tgx + tgw * 0.5f;
    float qy1 = tgy - tgh * 0.5f, qy2 = tgy + tgh * 0.5f;
    float iw = fmaxf(fminf(px2, qx2) - fmaxf(px1, qx1), 0.0f);
    float ih = fmaxf(fminf(py2, qy2) - fmaxf(py1, qy1), 0.0f);
    float inter = iw * ih;
    float iou_t = inter / (pw * ph + tgw * tgh - inter + 1e-16f);

    float pconf = sigf(cfv);
    float objSE = (pconf - iou_t) * (pconf - iou_t);

    float clsSE = 0.0f;
    for (int c = 0; c < NC; ++c) {
        float pc = sigf(x[base + (5 + c) * GG]);
        float on = (c == ci) ? 1.0f : 0.0f;
        clsSE += (pc - on) * (pc - on);
    }

    float sc = sqrtf(2.0f - tgw * tgh / (float)GG);
    float e0 = (sx - (tgx - (float)gx)) * sc;
    float e1 = (sy - (tgy - (float)gy)) * sc;
    float e2 = (twv - logf(tgw / c_AW[ai])) * sc;
    float e3 = (thv - logf(tgh / c_AH[ai])) * sc;
    float coordSE = e0 * e0 + e1 * e1 + e2 * e2 + e3 * e3;

    float cS = block_sum(clsSE, red);
    float oS = block_sum(objSE, red);
    float dS = block_sum(coordSE, red);
    if (threadIdx.x == 0) {
        scal[0] = cS / (float)(NT * NC);          // CLASS_SCALE = 1
        scal[1] = oS / (float)NT * 5.0f;          // OBJECT_SCALE = 5
        scal[2] = dS / (float)(NT * 4);           // COORD_SCALE = 1
    }
}

// ---- kernel 2: full-grid noobj / prior pass ------------------------------
__global__ void k_grid(const float* __restrict__ x,
                       const int* __restrict__ mask,
                       const float* __restrict__ tbox,
                       float* __restrict__ partials) {
    __shared__ float sX1[NT], sY1[NT], sX2[NT], sY2[NT], sAr[NT];
    __shared__ float red[8];
    int t = threadIdx.x;
    sX1[t] = tbox[0 * NT + t];
    sY1[t] = tbox[1 * NT + t];
    sX2[t] = tbox[2 * NT + t];
    sY2[t] = tbox[3 * NT + t];
    sAr[t] = tbox[4 * NT + t];
    __syncthreads();

    int idx = blockIdx.x * blockDim.x + threadIdx.x;
    float noobjV = 0.0f, ncntV = 0.0f, priorV = 0.0f, nmV = 0.0f;

    if (idx < NCELL) {
        int gx = idx % GDIM;
        int gy = (idx / GDIM) % GDIM;
        int a  = (idx / GG) % NA;
        int b  = idx / (NA * GG);

        int base = (b * NA * NCH + a * NCH) * GG + gy * GDIM + gx;
        float txv = x[base + 0 * GG], tyv = x[base + 1 * GG];
        float twv = x[base + 2 * GG], thv = x[base + 3 * GG];
        float cfv = x[base + 4 * GG];

        float sx = sigf(txv), sy = sigf(tyv), pconf = sigf(cfv);
        float pw = expf(twv) * c_AW[a], ph = expf(thv) * c_AH[a];
        float px = sx + (float)gx, py = sy + (float)gy;
        float px1 = px - pw * 0.5f, px2 = px + pw * 0.5f;
        float py1 = py - ph * 0.5f, py2 = py + ph * 0.5f;
        float parea = pw * ph;

        float mx = 0.0f;
        for (int k = 0; k < NT; ++k) {
            float iw = fmaxf(fminf(px2, sX2[k]) - fmaxf(px1, sX1[k]), 0.0f);
            float ih = fmaxf(fminf(py2, sY2[k]) - fmaxf(py1, sY1[k]), 0.0f);
            float inter = iw * ih;
            float iou = inter / (parea + sAr[k] - inter + 1e-16f);
            mx = fmaxf(mx, iou);
        }

        if (mask[idx] == 0) {
            nmV = 1.0f;
            float d0 = sx - 0.5f, d1 = sy - 0.5f;
            priorV = d0 * d0 + d1 * d1 + twv * twv + thv * thv;
            if (mx <= 0.6f) { ncntV = 1.0f; noobjV = pconf * pconf; }
        }
    }

    float s0 = block_sum(noobjV, red);
    float s1 = block_sum(ncntV, red);
    float s2 = block_sum(priorV, red);
    float s3 = block_sum(nmV, red);
    if (threadIdx.x == 0) {
        partials[blockIdx.x * 4 + 0] = s0;
        partials[blockIdx.x * 4 + 1] = s1;
        partials[blockIdx.x * 4 + 2] = s2;
        partials[blockIdx.x * 4 + 3] = s3;
    }
}

// ---- kernel 3: finalize --------------------------------------------------
__global__ void k_final(const float* __restrict__ scal,
                        const float* __restrict__ partials,
                        const int* __restrict__ seen,
                        float* __restrict__ out) {
    float no = 0.0f, nc = 0.0f, pr = 0.0f, nm = 0.0f;
    for (int i = 0; i < NBLK; ++i) {
        no += partials[i * 4 + 0];
        nc += partials[i * 4 + 1];
        pr += partials[i * 4 + 2];
        nm += partials[i * 4 + 3];
    }
    float total = scal[0] + scal[1] + scal[2] + no / nc;  // NOOBJECT_SCALE = 1
    if (seen[0] < 12800)
        total += pr / (nm * 4.0f) * 0.01f;
    out[0] = total;
}

extern "C" void kernel_launch(void* const* d_in, const int* in_sizes, int n_in,
                              void* d_out, int out_size, void* d_ws, size_t ws_size,
                              hipStream_t stream) {
    const float* x  = (const float*)d_in[0];
    const float* tg = (const float*)d_in[1];
    const int* seen = (const int*)d_in[2];
    float* out = (float*)d_out;

    int*   mask     = (int*)d_ws;                    // NCELL ints
    float* tbox     = (float*)d_ws + NCELL;          // 5*NT floats
    float* scal     = tbox + 5 * NT;                 // 3 floats (+pad)
    float* partials = scal + 8;                      // 4*NBLK floats

    k_zero_mask<<<NBLK, 256, 0, stream>>>(mask);
    k_targets  <<<1,    256, 0, stream>>>(x, tg, mask, tbox, scal);
    k_grid     <<<NBLK, 256, 0, stream>>>(x, mask, tbox, partials);
    k_final    <<<1,    1,   0, stream>>>(scal, partials, seen, out);
}